// TorchDiAcyclicGraph_51762945851844
// MI455X (gfx1250) — compile-verified
//
#include <hip/hip_runtime.h>

typedef float v2f __attribute__((ext_vector_type(2)));
typedef float v8f __attribute__((ext_vector_type(8)));

#define NUM_LAYERS  4
#define NUM_FOLDS   512
#define FOLD_DIM    64
#define BATCH       2048
#define NUM_OUTPUTS 8
#define BM          128   // batch rows per workgroup

// One layer: for fold f, H[b,:] = x[b,i0,:]*x[b,i1,:]; Y[b,f,:] = H[b,:] @ W[f]
// Per-block: fold f, batch rows [b0, b0+BM). 8 waves, each wave = one 16-row
// M tile x four 16-col N tiles, K=64 via 16 chained V_WMMA_F32_16X16X4_F32.
__global__ __launch_bounds__(256) void fold_layer_kernel(
    const float* __restrict__ src,   // [BATCH, NUM_FOLDS, FOLD_DIM]
    const float* __restrict__ wl,    // [NUM_FOLDS, 64, 64] (this layer)
    const int*   __restrict__ idxl,  // [NUM_FOLDS, 2]     (this layer)
    float* __restrict__ dst)         // [BATCH, NUM_FOLDS, FOLD_DIM]
{
    __shared__ float sH [BM * FOLD_DIM];        // A tile, row-major [m][k] (32 KB)
    __shared__ float sWT[FOLD_DIM * FOLD_DIM];  // B tile transposed [n][k] (16 KB)

    const int f   = blockIdx.x;
    const int b0  = blockIdx.y * BM;
    const int tid = threadIdx.x;

    const size_t rowstride = (size_t)NUM_FOLDS * FOLD_DIM;

    const int i0 = idxl[f * 2 + 0];
    const int i1 = idxl[f * 2 + 1];
    const float* x0 = src + (size_t)i0 * FOLD_DIM;
    const float* x1 = src + (size_t)i1 * FOLD_DIM;

    // --- Stage H = gather0 * gather1 into LDS (coalesced float4) -------------
    for (int t = tid; t < BM * (FOLD_DIM / 4); t += 256) {
        const int r = t >> 4;            // row 0..BM-1
        const int c = (t & 15) << 2;     // col 0..60
        const size_t g = (size_t)(b0 + r) * rowstride + c;
        float4 a = *(const float4*)(x0 + g);
        float4 b = *(const float4*)(x1 + g);
        float4 h = make_float4(a.x * b.x, a.y * b.y, a.z * b.z, a.w * b.w);
        *(float4*)&sH[r * FOLD_DIM + c] = h;
    }

    // --- Stage W[f] transposed: sWT[n][k] = W[k][n] --------------------------
    const float* wf = wl + (size_t)f * FOLD_DIM * FOLD_DIM;
    for (int t = tid; t < FOLD_DIM * (FOLD_DIM / 4); t += 256) {
        const int k = t >> 4;
        const int n = (t & 15) << 2;
        float4 w = *(const float4*)(wf + k * FOLD_DIM + n);
        sWT[(n + 0) * FOLD_DIM + k] = w.x;
        sWT[(n + 1) * FOLD_DIM + k] = w.y;
        sWT[(n + 2) * FOLD_DIM + k] = w.z;
        sWT[(n + 3) * FOLD_DIM + k] = w.w;
    }

    __syncthreads();

    // --- WMMA: wave w owns M-tile w; 4 N-tiles; K=64 in steps of 4 -----------
    const int wave   = tid >> 5;        // 0..7
    const int lane   = tid & 31;
    const int lane15 = lane & 15;
    const int khalf  = (lane >> 4) << 1;    // 0 (lanes 0-15) or 2 (lanes 16-31)

    // A frag: lane -> H[mtile*16 + lane15, kk + khalf + {0,1}]  (contiguous pair)
    const float* aRow  = &sH [(wave * 16 + lane15) * FOLD_DIM + khalf];
    // B frag: lane -> W[kk + khalf + {0,1}, ntile*16 + lane15] == WT[n][k..k+1]
    const float* bRow0 = &sWT[( 0 + lane15) * FOLD_DIM + khalf];
    const float* bRow1 = &sWT[(16 + lane15) * FOLD_DIM + khalf];
    const float* bRow2 = &sWT[(32 + lane15) * FOLD_DIM + khalf];
    const float* bRow3 = &sWT[(48 + lane15) * FOLD_DIM + khalf];

    v8f acc0 = {}, acc1 = {}, acc2 = {}, acc3 = {};

#pragma unroll
    for (int kk = 0; kk < FOLD_DIM; kk += 4) {
        v2f a  = *(const v2f*)(aRow  + kk);
        v2f b0v = *(const v2f*)(bRow0 + kk);
        v2f b1v = *(const v2f*)(bRow1 + kk);
        v2f b2v = *(const v2f*)(bRow2 + kk);
        v2f b3v = *(const v2f*)(bRow3 + kk);
        acc0 = __builtin_amdgcn_wmma_f32_16x16x4_f32(false, a, false, b0v,
                                                     (short)0, acc0, false, false);
        acc1 = __builtin_amdgcn_wmma_f32_16x16x4_f32(false, a, false, b1v,
                                                     (short)0, acc1, false, false);
        acc2 = __builtin_amdgcn_wmma_f32_16x16x4_f32(false, a, false, b2v,
                                                     (short)0, acc2, false, false);
        acc3 = __builtin_amdgcn_wmma_f32_16x16x4_f32(false, a, false, b3v,
                                                     (short)0, acc3, false, false);
    }

    // --- Store D: VGPR r -> rows r (lanes 0-15) / r+8 (lanes 16-31) ----------
    const int mofs = (lane >> 4) * 8;
    float* drow = dst + (size_t)(b0 + wave * 16 + mofs) * rowstride
                      + (size_t)f * FOLD_DIM + lane15;
#pragma unroll
    for (int r = 0; r < 8; ++r) {
        const size_t off = (size_t)r * rowstride;
        drow[off +  0] = acc0[r];
        drow[off + 16] = acc1[r];
        drow[off + 32] = acc2[r];
        drow[off + 48] = acc3[r];
    }
}

__global__ __launch_bounds__(256) void gather_out_kernel(
    const float* __restrict__ x,        // [BATCH, NUM_FOLDS, FOLD_DIM]
    const int*   __restrict__ out_idx,  // [NUM_OUTPUTS]
    float* __restrict__ out)            // [BATCH, NUM_OUTPUTS, FOLD_DIM]
{
    const int t = blockIdx.x * blockDim.x + threadIdx.x;
    const int total = BATCH * NUM_OUTPUTS * FOLD_DIM;
    if (t >= total) return;
    const int k = t & (FOLD_DIM - 1);
    const int o = (t >> 6) & (NUM_OUTPUTS - 1);
    const int b = t >> 9;
    out[t] = x[(size_t)b * NUM_FOLDS * FOLD_DIM
               + (size_t)out_idx[o] * FOLD_DIM + k];
}

extern "C" void kernel_launch(void* const* d_in, const int* in_sizes, int n_in,
                              void* d_out, int out_size, void* d_ws, size_t ws_size,
                              hipStream_t stream) {
    const float* in_graph = (const float*)d_in[0];  // (2048, 512, 64) f32
    const float* weights  = (const float*)d_in[1];  // (4, 512, 64, 64) f32
    const int*   fold_idx = (const int*)d_in[2];    // (4, 512, 2) i32
    const int*   out_idx  = (const int*)d_in[3];    // (8,) i32
    float* out = (float*)d_out;                     // (2048, 8, 64) f32

    const size_t XN = (size_t)BATCH * NUM_FOLDS * FOLD_DIM; // 64M floats
    float* bufA = (float*)d_ws;
    float* bufB = bufA + XN;

    const size_t WL = (size_t)NUM_FOLDS * FOLD_DIM * FOLD_DIM;
    const size_t IL = (size_t)NUM_FOLDS * 2;

    dim3 grid(NUM_FOLDS, BATCH / BM);
    dim3 block(256);

    const float* src = in_graph;
    float* dsts[NUM_LAYERS] = { bufA, bufB, bufA, bufB };
    for (int l = 0; l < NUM_LAYERS; ++l) {
        fold_layer_kernel<<<grid, block, 0, stream>>>(
            src, weights + (size_t)l * WL, fold_idx + (size_t)l * IL, dsts[l]);
        src = dsts[l];
    }

    const int n = BATCH * NUM_OUTPUTS * FOLD_DIM;
    gather_out_kernel<<<(n + 255) / 256, 256, 0, stream>>>(
        dsts[NUM_LAYERS - 1], out_idx, out);
}